// Gat_33337536151788
// MI455X (gfx1250) — compile-verified
//
#include <hip/hip_runtime.h>
#include <hip/hip_bf16.h>

// ---------------- problem constants (match reference) ----------------
#define NNODE   100000
#define NEDGE   1600000
#define NFEAT   256
#define HC      64            // 8 heads * 8 channels (layer-1 output width)
#define NHEAD   8
#define NCH     8
#define NCLASS  40
#define NEDGE_SL (NEDGE + NNODE)   // edges + self loops
#define NEG_SLOPE 0.2f
#define EPS_F   1e-16f

typedef float v2f __attribute__((ext_vector_type(2)));
typedef float v8f __attribute__((ext_vector_type(8)));

// ---------------- helpers ----------------
__device__ __forceinline__ v8f wmma_f32x4(v2f a, v2f b, v8f c) {
  // V_WMMA_F32_16X16X4_F32 : D(16x16 f32) = A(16x4 f32) * B(4x16 f32) + C
  return __builtin_amdgcn_wmma_f32_16x16x4_f32(
      /*neg_a=*/false, a, /*neg_b=*/false, b,
      /*c_mod=*/(short)0, c, /*reuse_a=*/false, /*reuse_b=*/false);
}

// monotonic float<->int key for atomicMax-based segment max
__device__ __forceinline__ int f2key(float f) {
  int i = __float_as_int(f);
  return i >= 0 ? i : (i ^ 0x7fffffff);
}
__device__ __forceinline__ float key2f(int k) {
  return __int_as_float(k >= 0 ? k : (k ^ 0x7fffffff));
}

__device__ __forceinline__ void edge_sd(const int* __restrict__ ei, long long e,
                                        int& s, int& d) {
  if (e < (long long)NEDGE) { s = ei[e]; d = ei[(long long)NEDGE + e]; }
  else { s = d = (int)(e - NEDGE); }           // virtual self loop
}

__device__ __forceinline__ float lrelu(float v) {
  return v > 0.f ? v : NEG_SLOPE * v;
}

// ---------------- layer 1 GEMM: H1[N,64] = X[N,256] @ W1[256,64] (WMMA f32) ---
__global__ __launch_bounds__(32) void gemm1_wmma(const float* __restrict__ X,
                                                 const float* __restrict__ W,
                                                 float* __restrict__ H) {
  const int lane = threadIdx.x & 31;
  const int half = lane >> 4;       // 0: K={0,1}, 1: K={2,3} of each k-step
  const int l15  = lane & 15;
  const int rowBase = blockIdx.x * 16;              // N = 6250 * 16 exactly
  const float* xrow = X + (size_t)(rowBase + l15) * NFEAT + 2 * half;

  v8f acc0 = {}, acc1 = {}, acc2 = {}, acc3 = {};
  for (int k = 0; k < NFEAT; k += 4) {
    __builtin_prefetch(xrow + k + 32, 0, 0);        // global_prefetch_b8
    v2f a; a.x = xrow[k]; a.y = xrow[k + 1];        // A 16x4 fragment
    const float* wr = W + (size_t)(k + 2 * half) * HC + l15;   // B 4x16 frags
    v2f b0; b0.x = wr[0];  b0.y = wr[HC + 0];
    v2f b1; b1.x = wr[16]; b1.y = wr[HC + 16];
    v2f b2; b2.x = wr[32]; b2.y = wr[HC + 32];
    v2f b3; b3.x = wr[48]; b3.y = wr[HC + 48];
    acc0 = wmma_f32x4(a, b0, acc0);
    acc1 = wmma_f32x4(a, b1, acc1);
    acc2 = wmma_f32x4(a, b2, acc2);
    acc3 = wmma_f32x4(a, b3, acc3);
  }
#pragma unroll
  for (int r = 0; r < 8; ++r) {                    // C/D: VGPR r -> rows r, r+8
    int row = rowBase + r + 8 * half;
    float* hp = H + (size_t)row * HC + l15;
    hp[0]  = acc0[r];
    hp[16] = acc1[r];
    hp[32] = acc2[r];
    hp[48] = acc3[r];
  }
}

// ---------------- layer 2 GEMM: H2[N,40] = X2[N,64] @ W2[64,40] (WMMA f32) ----
__global__ __launch_bounds__(32) void gemm2_wmma(const float* __restrict__ X2,
                                                 const float* __restrict__ W,
                                                 float* __restrict__ H) {
  const int lane = threadIdx.x & 31;
  const int half = lane >> 4;
  const int l15  = lane & 15;
  const int rowBase = blockIdx.x * 16;

  v8f acc0 = {}, acc1 = {}, acc2 = {};
  const float* xrow = X2 + (size_t)(rowBase + l15) * HC + 2 * half;
  for (int k = 0; k < HC; k += 4) {
    v2f a; a.x = xrow[k]; a.y = xrow[k + 1];
    const int kr = k + 2 * half;
    v2f b0, b1, b2;
    {
      int n = l15;          // col tile 0: cols 0..15 (all < 40)
      b0.x = W[(size_t)kr * NCLASS + n];
      b0.y = W[(size_t)(kr + 1) * NCLASS + n];
    }
    {
      int n = 16 + l15;     // col tile 1: cols 16..31
      b1.x = W[(size_t)kr * NCLASS + n];
      b1.y = W[(size_t)(kr + 1) * NCLASS + n];
    }
    {
      int n = 32 + l15;     // col tile 2: cols 32..47, pad >= 40 with zero
      bool ok = n < NCLASS;
      b2.x = ok ? W[(size_t)kr * NCLASS + n] : 0.f;
      b2.y = ok ? W[(size_t)(kr + 1) * NCLASS + n] : 0.f;
    }
    acc0 = wmma_f32x4(a, b0, acc0);
    acc1 = wmma_f32x4(a, b1, acc1);
    acc2 = wmma_f32x4(a, b2, acc2);
  }
#pragma unroll
  for (int r = 0; r < 8; ++r) {
    int row = rowBase + r + 8 * half;
    float* hp = H + (size_t)row * NCLASS;
    hp[l15]      = acc0[r];
    hp[16 + l15] = acc1[r];
    if (32 + l15 < NCLASS) hp[32 + l15] = acc2[r];
  }
}

// ---------------- attention coefficients, layer 1 ----------------
__global__ void attn_coef_l1(const float* __restrict__ H1,
                             const float* __restrict__ attS,
                             const float* __restrict__ attD,
                             float* __restrict__ aS, float* __restrict__ aD,
                             int* __restrict__ maxk) {
  int node = blockIdx.x * blockDim.x + threadIdx.x;
  if (node >= NNODE) return;
  const float* h = H1 + (size_t)node * HC;
#pragma unroll
  for (int hd = 0; hd < NHEAD; ++hd) {
    float s = 0.f, d = 0.f;
#pragma unroll
    for (int c = 0; c < NCH; ++c) {
      float v = h[hd * NCH + c];
      s += v * attS[hd * NCH + c];
      d += v * attD[hd * NCH + c];
    }
    aS[(size_t)node * NHEAD + hd] = s;
    aD[(size_t)node * NHEAD + hd] = d;
    maxk[(size_t)node * NHEAD + hd] = f2key(-INFINITY);
  }
}

// ---------------- layer-1 edge passes (thread per edge*head) ----------------
__global__ void edge_max_l1(const int* __restrict__ ei,
                            const float* __restrict__ aS,
                            const float* __restrict__ aD,
                            float* __restrict__ e1, int* __restrict__ maxk) {
  long long t = (long long)blockIdx.x * blockDim.x + threadIdx.x;
  if (t >= (long long)NEDGE_SL * NHEAD) return;
  long long e = t >> 3; int hd = (int)(t & 7);
  int s, d; edge_sd(ei, e, s, d);
  float v = lrelu(aS[(size_t)s * NHEAD + hd] + aD[(size_t)d * NHEAD + hd]);
  e1[t] = v;
  atomicMax(&maxk[(size_t)d * NHEAD + hd], f2key(v));
}

__global__ void edge_sum_l1(const int* __restrict__ ei,
                            const float* __restrict__ e1,
                            const int* __restrict__ maxk,
                            float* __restrict__ sum) {
  long long t = (long long)blockIdx.x * blockDim.x + threadIdx.x;
  if (t >= (long long)NEDGE_SL * NHEAD) return;
  long long e = t >> 3; int hd = (int)(t & 7);
  int s, d; edge_sd(ei, e, s, d);
  float m = key2f(maxk[(size_t)d * NHEAD + hd]);
  atomicAdd(&sum[(size_t)d * NHEAD + hd], __expf(e1[t] - m));
}

__global__ void edge_scatter_l1(const int* __restrict__ ei,
                                const float* __restrict__ e1,
                                const int* __restrict__ maxk,
                                const float* __restrict__ sum,
                                const float* __restrict__ H1,
                                float* __restrict__ out1) {
  long long t = (long long)blockIdx.x * blockDim.x + threadIdx.x;
  if (t >= (long long)NEDGE_SL * NHEAD) return;
  long long e = t >> 3; int hd = (int)(t & 7);
  int s, d; edge_sd(ei, e, s, d);
  size_t d8 = (size_t)d * NHEAD + hd;
  float alpha = __expf(e1[t] - key2f(maxk[d8])) / (sum[d8] + EPS_F);
  const float* hs = H1 + (size_t)s * HC + hd * NCH;
  float* op = out1 + (size_t)d * HC + hd * NCH;
#pragma unroll
  for (int c = 0; c < NCH; ++c) atomicAdd(&op[c], hs[c] * alpha);
}

// ---------------- finalize layer 1: +b1 then ELU (in place -> X2) ------------
__global__ void node_bias_elu(float* __restrict__ out1,
                              const float* __restrict__ b1) {
  long long i = (long long)blockIdx.x * blockDim.x + threadIdx.x;
  if (i >= (long long)NNODE * HC) return;
  float v = out1[i] + b1[(int)(i & (HC - 1))];
  out1[i] = v > 0.f ? v : (__expf(v) - 1.f);
}

// ---------------- attention coefficients, layer 2 (1 head, 40 ch) -----------
__global__ void attn_coef_l2(const float* __restrict__ H2,
                             const float* __restrict__ attS,
                             const float* __restrict__ attD,
                             float* __restrict__ aS, float* __restrict__ aD,
                             int* __restrict__ maxk) {
  int node = blockIdx.x * blockDim.x + threadIdx.x;
  if (node >= NNODE) return;
  const float* h = H2 + (size_t)node * NCLASS;
  float s = 0.f, d = 0.f;
#pragma unroll
  for (int c = 0; c < NCLASS; ++c) {
    float v = h[c];
    s += v * attS[c];
    d += v * attD[c];
  }
  aS[node] = s; aD[node] = d; maxk[node] = f2key(-INFINITY);
}

// ---------------- layer-2 edge passes (thread per edge) ----------------
__global__ void edge_max_l2(const int* __restrict__ ei,
                            const float* __restrict__ aS,
                            const float* __restrict__ aD,
                            float* __restrict__ e2, int* __restrict__ maxk) {
  long long e = (long long)blockIdx.x * blockDim.x + threadIdx.x;
  if (e >= (long long)NEDGE_SL) return;
  int s, d; edge_sd(ei, e, s, d);
  float v = lrelu(aS[s] + aD[d]);
  e2[e] = v;
  atomicMax(&maxk[d], f2key(v));
}

__global__ void edge_sum_l2(const int* __restrict__ ei,
                            const float* __restrict__ e2,
                            const int* __restrict__ maxk,
                            float* __restrict__ sum) {
  long long e = (long long)blockIdx.x * blockDim.x + threadIdx.x;
  if (e >= (long long)NEDGE_SL) return;
  int s, d; edge_sd(ei, e, s, d);
  atomicAdd(&sum[d], __expf(e2[e] - key2f(maxk[d])));
}

// thread per (edge, 8-channel chunk): 40 = 5 chunks
__global__ void edge_scatter_l2(const int* __restrict__ ei,
                                const float* __restrict__ e2,
                                const int* __restrict__ maxk,
                                const float* __restrict__ sum,
                                const float* __restrict__ H2,
                                float* __restrict__ out) {
  long long t = (long long)blockIdx.x * blockDim.x + threadIdx.x;
  if (t >= (long long)NEDGE_SL * 5) return;
  long long e = t / 5; int chunk = (int)(t - e * 5) * 8;
  int s, d; edge_sd(ei, e, s, d);
  float alpha = __expf(e2[e] - key2f(maxk[d])) / (sum[d] + EPS_F);
  const float* hs = H2 + (size_t)s * NCLASS + chunk;
  float* op = out + (size_t)d * NCLASS + chunk;
#pragma unroll
  for (int c = 0; c < 8; ++c) atomicAdd(&op[c], hs[c] * alpha);
}

// ---------------- final bias ----------------
__global__ void out_bias(float* __restrict__ out, const float* __restrict__ b2) {
  long long i = (long long)blockIdx.x * blockDim.x + threadIdx.x;
  if (i >= (long long)NNODE * NCLASS) return;
  int c = (int)(i - (i / NCLASS) * NCLASS);
  out[i] += b2[c];
}

// ---------------- launch ----------------
extern "C" void kernel_launch(void* const* d_in, const int* in_sizes, int n_in,
                              void* d_out, int out_size, void* d_ws, size_t ws_size,
                              hipStream_t stream) {
  const float* x   = (const float*)d_in[0];
  const int*   ei  = (const int*)d_in[1];     // [2,E] flat: src then dst
  const float* W1  = (const float*)d_in[2];
  const float* as1 = (const float*)d_in[3];
  const float* ad1 = (const float*)d_in[4];
  const float* b1  = (const float*)d_in[5];
  const float* W2  = (const float*)d_in[6];
  const float* as2 = (const float*)d_in[7];
  const float* ad2 = (const float*)d_in[8];
  const float* b2  = (const float*)d_in[9];
  float* out = (float*)d_out;

  // ---- workspace layout (floats) ----
  float* ws = (float*)d_ws;
  size_t o = 0;
  float* H1  = ws + o; o += (size_t)NNODE * HC;        // 6.4 M
  float* aS1 = ws + o; o += (size_t)NNODE * NHEAD;
  float* aD1 = ws + o; o += (size_t)NNODE * NHEAD;
  int*   mx1 = (int*)(ws + o); o += (size_t)NNODE * NHEAD;
  float* sm1 = ws + o; o += (size_t)NNODE * NHEAD;
  float* e1  = ws + o; o += (size_t)NEDGE_SL * NHEAD;  // 13.6 M
  float* X2  = ws + o; o += (size_t)NNODE * HC;        // out1 accum -> ELU'd
  float* H2  = ws + o; o += (size_t)NNODE * NCLASS;
  float* aS2 = ws + o; o += (size_t)NNODE;
  float* aD2 = ws + o; o += (size_t)NNODE;
  int*   mx2 = (int*)(ws + o); o += (size_t)NNODE;
  float* sm2 = ws + o; o += (size_t)NNODE;
  float* e2  = ws + o; o += (size_t)NEDGE_SL;

  // ---- zero the accumulators (graph-capture-safe async memsets) ----
  hipMemsetAsync(X2,  0, (size_t)NNODE * HC * sizeof(float), stream);
  hipMemsetAsync(sm1, 0, (size_t)NNODE * NHEAD * sizeof(float), stream);
  hipMemsetAsync(sm2, 0, (size_t)NNODE * sizeof(float), stream);
  hipMemsetAsync(out, 0, (size_t)NNODE * NCLASS * sizeof(float), stream);

  const int TB = 256;
  const long long EH1 = (long long)NEDGE_SL * NHEAD;
  const long long E2C = (long long)NEDGE_SL * 5;

  // ---- layer 1 ----
  gemm1_wmma<<<NNODE / 16, 32, 0, stream>>>(x, W1, H1);
  attn_coef_l1<<<(NNODE + 127) / 128, 128, 0, stream>>>(H1, as1, ad1, aS1, aD1, mx1);
  edge_max_l1<<<(int)((EH1 + TB - 1) / TB), TB, 0, stream>>>(ei, aS1, aD1, e1, mx1);
  edge_sum_l1<<<(int)((EH1 + TB - 1) / TB), TB, 0, stream>>>(ei, e1, mx1, sm1);
  edge_scatter_l1<<<(int)((EH1 + TB - 1) / TB), TB, 0, stream>>>(ei, e1, mx1, sm1, H1, X2);
  node_bias_elu<<<(int)(((long long)NNODE * HC + TB - 1) / TB), TB, 0, stream>>>(X2, b1);

  // ---- layer 2 ----
  gemm2_wmma<<<NNODE / 16, 32, 0, stream>>>(X2, W2, H2);
  attn_coef_l2<<<(NNODE + 127) / 128, 128, 0, stream>>>(H2, as2, ad2, aS2, aD2, mx2);
  edge_max_l2<<<(int)(((long long)NEDGE_SL + TB - 1) / TB), TB, 0, stream>>>(ei, aS2, aD2, e2, mx2);
  edge_sum_l2<<<(int)(((long long)NEDGE_SL + TB - 1) / TB), TB, 0, stream>>>(ei, e2, mx2, sm2);
  edge_scatter_l2<<<(int)((E2C + TB - 1) / TB), TB, 0, stream>>>(ei, e2, mx2, sm2, H2, out);
  out_bias<<<(int)(((long long)NNODE * NCLASS + TB - 1) / TB), TB, 0, stream>>>(out, b2);
}